// Model_71270687310451
// MI455X (gfx1250) — compile-verified
//
#include <hip/hip_runtime.h>

// ---------------------------------------------------------------------------
// Model: GCN(node_emb gather) -> tanh -> +class_emb -> plm proj -> tanh
//        -> W1/relu -> W2 reduce.  Word-embedding branch is dead code
//        (output only consumes merge rows 1..50 == GCN features).
// GEMMs: V_WMMA_F32_16X16X4_F32.  A-tile staging: GLOBAL_LOAD_ASYNC_TO_LDS_B128
// (ASYNCcnt path) instead of global->VGPR->LDS round trip.
// ---------------------------------------------------------------------------

#define DDIM   256
#define BATCH  32
#define NNODES 256
#define NEDGES 1024
#define CLS    50
#define HCLS   1024

typedef float v2f __attribute__((ext_vector_type(2)));
typedef float v4f __attribute__((ext_vector_type(4)));
typedef float v8f __attribute__((ext_vector_type(8)));

// Direct global->LDS async copy of 16B per lane.  Flat LDS pointers carry the
// wave-relative LDS offset in their low 32 bits (ISA 10.2), which is what the
// VDST operand of global_load_async_to_lds_* expects.  INST_OFFSET is added to
// both the LDS and the global address (ISA 08 4.4).
__device__ __forceinline__ void async_cp16(void* lds_dst, const void* gsrc, int imm_off)
{
    unsigned lds = (unsigned)(size_t)lds_dst;
    switch (imm_off) {   // immediate must be a literal in the asm string
    case 0:  asm volatile("global_load_async_to_lds_b128 %0, %1, off"
                          :: "v"(lds), "v"(gsrc) : "memory"); break;
    case 16: asm volatile("global_load_async_to_lds_b128 %0, %1, off offset:16"
                          :: "v"(lds), "v"(gsrc) : "memory"); break;
    case 32: asm volatile("global_load_async_to_lds_b128 %0, %1, off offset:32"
                          :: "v"(lds), "v"(gsrc) : "memory"); break;
    default: asm volatile("global_load_async_to_lds_b128 %0, %1, off offset:48"
                          :: "v"(lds), "v"(gsrc) : "memory"); break;
    }
}
__device__ __forceinline__ void async_wait0()
{
    asm volatile("s_wait_asynccnt 0" ::: "memory");
}

// 16x16 fp32 WMMA tile over K=256.
// Layout (ISA 7.12.2, 32-bit): A lane=16*half+m : VGPR0=A[m][2*half], VGPR1=A[m][2*half+1]
//                              B lane=16*half+n : VGPR0=B[2*half][n], VGPR1=B[2*half+1][n]
//                              C/D lane=16*half+n, VGPR v = C[v+8*half][n]

// --- K1: h[g][:] = node_emb[nodes[g]] @ gcn_W, g = b*256+n (8192 rows) -----
__global__ __launch_bounds__(256) void k_gcn_mm(const int* __restrict__ nodes,
                                                const float* __restrict__ node_emb,
                                                const float* __restrict__ gcn_W,
                                                float* __restrict__ h)
{
    __shared__ float As[16][DDIM];
    const int tid = threadIdx.x;
    const int row0 = blockIdx.x * 16;
    {   // gather 16 embedding rows straight into LDS (64B per thread, async)
        const int l = tid >> 4, seg = tid & 15;
        const int g = row0 + l;
        const int b = g >> 8, nl = g & 255;
        const long node = nodes[b * NNODES + nl];
        const float* src = node_emb + node * DDIM + seg * 16;
        void* dst = &As[l][seg * 16];
        async_cp16(dst, src, 0);
        async_cp16(dst, src, 16);
        async_cp16(dst, src, 32);
        async_cp16(dst, src, 48);
    }
    async_wait0();
    __syncthreads();

    const int wave = tid >> 5, lane = tid & 31;
    const int half = lane >> 4, mrow = lane & 15, ncl = lane & 15;

    for (int t = 0; t < 2; ++t) {          // 8 waves x 2 = 16 column tiles
        const int ncol = (wave * 2 + t) * 16 + ncl;
        v8f acc = {};
        for (int kk = 0; kk < 64; ++kk) {
            const int kA = kk * 4 + half * 2;
            v2f a, bv;
            a.x  = As[mrow][kA];
            a.y  = As[mrow][kA + 1];
            bv.x = gcn_W[kA * DDIM + ncol];
            bv.y = gcn_W[(kA + 1) * DDIM + ncol];
            acc = __builtin_amdgcn_wmma_f32_16x16x4_f32(false, a, false, bv,
                                                        (short)0, acc, false, false);
        }
        #pragma unroll
        for (int v = 0; v < 8; ++v)
            h[(long)(row0 + v + half * 8) * DDIM + ncol] = acc[v];
    }
}

// --- K2: normalized aggregation + tanh + class_emb (rows 0..49 only) -------
__global__ __launch_bounds__(256) void k_gcn_agg(const int* __restrict__ edges,
                                                 const float* __restrict__ h,
                                                 const float* __restrict__ gcn_b,
                                                 const float* __restrict__ class_emb,
                                                 float* __restrict__ features)
{
    __shared__ float acc[CLS][DDIM];
    __shared__ float dinv[NNODES];
    __shared__ int   deg[NNODES];
    const int b = blockIdx.x, tid = threadIdx.x;
    const int* src = edges + (long)b * 2 * NEDGES;
    const int* dst = src + NEDGES;
    const float* hb = h + (long)b * NNODES * DDIM;

    deg[tid] = 1;                       // self loop
    __syncthreads();
    for (int e = tid; e < NEDGES; e += 256) atomicAdd(&deg[dst[e]], 1);
    __syncthreads();
    dinv[tid] = rsqrtf((float)deg[tid]);
    for (int c = 0; c < CLS; ++c) acc[c][tid] = 0.0f;
    __syncthreads();

    // thread `tid` owns feature column `tid` -> race-free scatter, no atomics
    for (int e = 0; e < NEDGES; ++e) {
        const int s = src[e], t = dst[e];
        if (t < CLS)
            acc[t][tid] += dinv[s] * dinv[t] * hb[s * DDIM + tid];
    }
    for (int c = 0; c < CLS; ++c) {
        const float val = acc[c][tid] + dinv[c] * dinv[c] * hb[c * DDIM + tid] + gcn_b[tid];
        features[((long)b * CLS + c) * DDIM + tid] = tanhf(val) + class_emb[c * DDIM + tid];
    }
}

// --- K3: t = tanh(features @ plm_W + plm_b), M padded 50->64 ---------------
__global__ __launch_bounds__(256) void k_plm(const float* __restrict__ features,
                                             const float* __restrict__ plm_W,
                                             const float* __restrict__ plm_b,
                                             float* __restrict__ tbuf)
{
    __shared__ float As[16][DDIM];
    const int tid = threadIdx.x;
    const int b = blockIdx.x >> 2, m_tile = blockIdx.x & 3;
    {
        const int l = tid >> 4, seg = tid & 15;
        const int c = m_tile * 16 + l;
        if (c < CLS) {       // wave-uniform branch (l is uniform per wave-half pair)
            const float* src = features + ((long)b * CLS + c) * DDIM + seg * 16;
            void* dstp = &As[l][seg * 16];
            async_cp16(dstp, src, 0);
            async_cp16(dstp, src, 16);
            async_cp16(dstp, src, 32);
            async_cp16(dstp, src, 48);
        } else {             // pad rows 50..63 with zeros
            v4f z = {};
            v4f* dp = (v4f*)&As[l][seg * 16];
            dp[0] = z; dp[1] = z; dp[2] = z; dp[3] = z;
        }
    }
    async_wait0();
    __syncthreads();

    const int wave = tid >> 5, lane = tid & 31;
    const int half = lane >> 4, mrow = lane & 15, ncl = lane & 15;

    for (int t = 0; t < 2; ++t) {
        const int ncol = (wave * 2 + t) * 16 + ncl;
        v8f acc = {};
        for (int kk = 0; kk < 64; ++kk) {
            const int kA = kk * 4 + half * 2;
            v2f a, bv;
            a.x  = As[mrow][kA];
            a.y  = As[mrow][kA + 1];
            bv.x = plm_W[kA * DDIM + ncol];
            bv.y = plm_W[(kA + 1) * DDIM + ncol];
            acc = __builtin_amdgcn_wmma_f32_16x16x4_f32(false, a, false, bv,
                                                        (short)0, acc, false, false);
        }
        const float bias = plm_b[ncol];
        #pragma unroll
        for (int v = 0; v < 8; ++v) {
            const int c = m_tile * 16 + v + half * 8;
            if (c < CLS)
                tbuf[((long)b * CLS + c) * DDIM + ncol] = tanhf(acc[v] + bias);
        }
    }
}

// --- K4: y[b,c] = sum_h W2[c,h] * relu(t[b,c,:] . W1[h,:] + b1[h]) ---------
__global__ __launch_bounds__(256) void k_cls(const float* __restrict__ tbuf,
                                             const float* __restrict__ W1,
                                             const float* __restrict__ b1,
                                             const float* __restrict__ W2,
                                             float* __restrict__ out)
{
    __shared__ float yacc[64];
    const int b = blockIdx.x, tid = threadIdx.x;
    if (tid < 64) yacc[tid] = 0.0f;
    __syncthreads();
    const float* tb = tbuf + (long)b * CLS * DDIM;

    const int wave = tid >> 5, lane = tid & 31;
    const int half = lane >> 4, mrow = lane & 15, ncl = lane & 15;

    // 4 M-tiles x 64 N-tiles over [64 x 1024] hidden, 8 waves round-robin
    for (int idx = wave; idx < 4 * 64; idx += 8) {
        const int m_tile = idx & 3, n_tile = idx >> 2;
        const int rowA = m_tile * 16 + mrow;
        const int ncol = n_tile * 16 + ncl;
        const bool avalid = rowA < CLS;
        v8f acc = {};
        for (int kk = 0; kk < 64; ++kk) {
            const int kA = kk * 4 + half * 2;
            v2f a, bv;
            a.x  = avalid ? tb[rowA * DDIM + kA]     : 0.0f;
            a.y  = avalid ? tb[rowA * DDIM + kA + 1] : 0.0f;
            bv.x = W1[ncol * DDIM + kA];        // W1^T access: B[k][n] = W1[n][k]
            bv.y = W1[ncol * DDIM + kA + 1];
            acc = __builtin_amdgcn_wmma_f32_16x16x4_f32(false, a, false, bv,
                                                        (short)0, acc, false, false);
        }
        const float bias = b1[ncol];
        #pragma unroll
        for (int v = 0; v < 8; ++v) {
            const int c = m_tile * 16 + v + half * 8;
            if (c < CLS) {
                const float hv = fmaxf(acc[v] + bias, 0.0f);
                atomicAdd(&yacc[c], W2[c * HCLS + ncol] * hv);   // ds_add_f32
            }
        }
    }
    __syncthreads();
    if (tid < CLS) out[b * CLS + tid] = yacc[tid];
}

// ---------------------------------------------------------------------------
extern "C" void kernel_launch(void* const* d_in, const int* in_sizes, int n_in,
                              void* d_out, int out_size, void* d_ws, size_t ws_size,
                              hipStream_t stream)
{
    (void)in_sizes; (void)n_in; (void)out_size; (void)ws_size;
    // setup_inputs() order
    const int*   nodes     = (const int*)  d_in[2];
    const int*   edges     = (const int*)  d_in[3];
    const float* node_emb  = (const float*)d_in[6];
    const float* class_emb = (const float*)d_in[8];
    const float* gcn_W     = (const float*)d_in[9];
    const float* gcn_b     = (const float*)d_in[10];
    const float* plm_W     = (const float*)d_in[11];
    const float* plm_b     = (const float*)d_in[12];
    const float* W1        = (const float*)d_in[13];
    const float* b1        = (const float*)d_in[14];
    const float* W2        = (const float*)d_in[15];
    float* out = (float*)d_out;

    float* ws       = (float*)d_ws;
    float* h        = ws;                                   // 32*256*256 (8 MB)
    float* features = h + (size_t)BATCH * NNODES * DDIM;    // 32*50*256 (1.6 MB)
    float* tbuf     = features + (size_t)BATCH * CLS * DDIM;// 32*50*256 (1.6 MB)

    k_gcn_mm<<<BATCH * NNODES / 16, 256, 0, stream>>>(nodes, node_emb, gcn_W, h);
    k_gcn_agg<<<BATCH, 256, 0, stream>>>(edges, h, gcn_b, class_emb, features);
    k_plm<<<BATCH * 4, 256, 0, stream>>>(features, plm_W, plm_b, tbuf);
    k_cls<<<BATCH, 256, 0, stream>>>(tbuf, W1, b1, W2, out);
}